// GraphAttention_10385230921846
// MI455X (gfx1250) — compile-verified
//
#include <hip/hip_runtime.h>
#include <hip/hip_bf16.h>

typedef __attribute__((ext_vector_type(2))) float v2f;
typedef __attribute__((ext_vector_type(4))) float v4f;
typedef __attribute__((ext_vector_type(8))) float v8f;

#define V_ 17
#define C_ 64
#define N_ 128
#define T_ 256
#define P_ 2
#define PITCH 68   // floats; (4*row + k) % 64 -> conflict-free b64 LDS fragment reads
#define ROWS 64    // output rows per block

// ---------------------------------------------------------------------------
// Prologue: fold softmax-attention column sums + BN into per-(v,n) scale and
// per-n bias.  a[l][n] = sum_m softmax_n(mat[m][l] * mask[m][l][:]).
//   S[l][n] = a[l][n] * gamma[n] / sqrt(var[n]+eps)
//   Bias[n] = beta[n] - mean[n]  * gamma[n] / sqrt(var[n]+eps)
// grid = (V_), block = (N_)
// ---------------------------------------------------------------------------
__global__ __launch_bounds__(N_) void ga_prologue(
    const float* __restrict__ mat, const float* __restrict__ amask,
    const float* __restrict__ gamma, const float* __restrict__ beta,
    const float* __restrict__ mean, const float* __restrict__ var,
    float* __restrict__ S, float* __restrict__ Bias)
{
    __shared__ float red[N_];
    const int l = blockIdx.x;
    const int n = threadIdx.x;

    float a_ln = 0.0f;
    for (int m = 0; m < V_; ++m) {
        const float z = mat[m * V_ + l] * amask[(m * V_ + l) * N_ + n];
        // max-reduce for numerical stability
        red[n] = z; __syncthreads();
        for (int s = N_ / 2; s > 0; s >>= 1) {
            if (n < s) red[n] = fmaxf(red[n], red[n + s]);
            __syncthreads();
        }
        const float mx = red[0]; __syncthreads();
        const float e = __expf(z - mx);
        red[n] = e; __syncthreads();
        for (int s = N_ / 2; s > 0; s >>= 1) {
            if (n < s) red[n] += red[n + s];
            __syncthreads();
        }
        const float sm = red[0]; __syncthreads();
        a_ln += e / sm;
    }

    const float inv = gamma[n] * rsqrtf(var[n] + 1e-3f);
    S[l * N_ + n] = a_ln * inv;
    if (l == 0) Bias[n] = beta[n] - mean[n] * inv;
}

// ---------------------------------------------------------------------------
// Main: per joint v, GEMM  y(BT x 64) * kern_v(64 x 128)  with f32 WMMA,
// fused scale/bias/relu epilogue, output layout (B,T,V,N).
// Streaming traffic (inputs/output, ~285 MB) uses non-temporal hints so the
// L2-resident kernel-weight slice (557 KB, re-read by every block) survives.
// grid = (BT/ROWS, V_), block = 256 threads (8 wave32)
// ---------------------------------------------------------------------------
__global__ __launch_bounds__(256) void ga_main(
    const float* __restrict__ in, const float* __restrict__ kern,
    const float* __restrict__ S, const float* __restrict__ Bias,
    float* __restrict__ out)
{
    __shared__ float As[ROWS * PITCH];   // [row][c]  (persons summed)
    __shared__ float Bs[N_ * PITCH];     // [n][c]    (kernel slice, transposed)

    const int tid = threadIdx.x;
    const int v = blockIdx.y;
    const int rowBase = blockIdx.x * ROWS;

    // ---- stage kernel slice: kern[(v*C_+c)*N_ + n]  ->  Bs[n][c]  (RT: cached) ----
    #pragma unroll
    for (int i = 0; i < 8; ++i) {
        const int e = tid + 256 * i;        // float4 unit, 0..2047
        const int c = e >> 5;               // 32 float4 per c-row (N_=128)
        const int n0 = (e & 31) << 2;
        const v4f kv = *(const v4f*)(kern + ((v * C_ + c) * N_ + n0));
        Bs[(n0 + 0) * PITCH + c] = kv.x;
        Bs[(n0 + 1) * PITCH + c] = kv.y;
        Bs[(n0 + 2) * PITCH + c] = kv.z;
        Bs[(n0 + 3) * PITCH + c] = kv.w;
    }

    // ---- stage A tile: sum over persons, coalesced non-temporal b128 loads ----
    {
        const int i = tid >> 2;             // row in tile (0..63)
        const int seg = tid & 3;
        const int M = rowBase + i;          // global (b*T + t)
        const int b = M / T_;
        const int t = M % T_;
        const int base0 = (((b * P_ + 0) * T_ + t) * V_ + v) * C_;
        const int base1 = base0 + T_ * V_ * C_;   // person 1
        #pragma unroll
        for (int s = 0; s < 4; ++s) {
            const int c = seg * 16 + s * 4;
            const v4f x0 = __builtin_nontemporal_load((const v4f*)(in + base0 + c));
            const v4f x1 = __builtin_nontemporal_load((const v4f*)(in + base1 + c));
            *(v4f*)(&As[i * PITCH + c]) = x0 + x1;   // (i*68+c)*4 is 16B aligned
        }
    }
    __syncthreads();

    const int lane = tid & 31;
    const int w = tid >> 5;
    const int half = lane >> 4;         // 0: lanes 0-15, 1: lanes 16-31
    const int mrow = lane & 15;
    const int rBase = (w & 3) * 16;     // wave's row offset within tile
    const int nBase = (w >> 2) * 64;    // wave's column offset

    v8f acc0 = {}, acc1 = {}, acc2 = {}, acc3 = {};

    // K = 64 in 16 steps of 4.  ISA A 16x4 layout: lanes 0-15 hold K={k0,k0+1},
    // lanes 16-31 hold K={k0+2,k0+3}; B 4x16 mirrors it with N across lanes.
    #pragma unroll
    for (int kk = 0; kk < 16; ++kk) {
        const int k0 = kk * 4 + half * 2;
        const v2f a  = *(const v2f*)(&As[(rBase + mrow) * PITCH + k0]);
        const v2f b0 = *(const v2f*)(&Bs[(nBase +  0 + mrow) * PITCH + k0]);
        const v2f b1 = *(const v2f*)(&Bs[(nBase + 16 + mrow) * PITCH + k0]);
        const v2f b2 = *(const v2f*)(&Bs[(nBase + 32 + mrow) * PITCH + k0]);
        const v2f b3 = *(const v2f*)(&Bs[(nBase + 48 + mrow) * PITCH + k0]);
        acc0 = __builtin_amdgcn_wmma_f32_16x16x4_f32(false, a, false, b0, (short)0, acc0, false, false);
        acc1 = __builtin_amdgcn_wmma_f32_16x16x4_f32(false, a, false, b1, (short)0, acc1, false, false);
        acc2 = __builtin_amdgcn_wmma_f32_16x16x4_f32(false, a, false, b2, (short)0, acc2, false, false);
        acc3 = __builtin_amdgcn_wmma_f32_16x16x4_f32(false, a, false, b3, (short)0, acc3, false, false);
    }

    // ---- epilogue: x*S[v,n] + Bias[n], relu, non-temporal store to (B,T,V,N) ----
    #pragma unroll
    for (int t = 0; t < 4; ++t) {
        const int n = nBase + t * 16 + mrow;
        const float sc = S[v * N_ + n];
        const float bi = Bias[n];
        const v8f acc = (t == 0) ? acc0 : (t == 1) ? acc1 : (t == 2) ? acc2 : acc3;
        #pragma unroll
        for (int r = 0; r < 8; ++r) {
            // C/D layout: VGPR r -> M = r (lanes 0-15) / r+8 (lanes 16-31)
            const int M = rowBase + rBase + r + half * 8;
            const float val = fmaxf(acc[r] * sc + bi, 0.0f);
            __builtin_nontemporal_store(val, out + (M * V_ + v) * N_ + n);
        }
    }
}

extern "C" void kernel_launch(void* const* d_in, const int* in_sizes, int n_in,
                              void* d_out, int out_size, void* d_ws, size_t ws_size,
                              hipStream_t stream) {
    const float* in    = (const float*)d_in[0];   // (B,P,T,V,C)
    const float* mat   = (const float*)d_in[1];   // (V,V)
    const float* amask = (const float*)d_in[2];   // (V,V,N)
    const float* kern  = (const float*)d_in[3];   // (V,C,N)
    const float* gamma = (const float*)d_in[4];
    const float* beta  = (const float*)d_in[5];
    const float* mean  = (const float*)d_in[6];
    const float* var   = (const float*)d_in[7];
    float* out = (float*)d_out;

    float* S    = (float*)d_ws;          // V_*N_ floats
    float* Bias = S + V_ * N_;           // N_ floats

    const int Mtotal = in_sizes[0] / (P_ * V_ * C_);   // B*T = 16384

    ga_prologue<<<dim3(V_), dim3(N_), 0, stream>>>(mat, amask, gamma, beta, mean, var, S, Bias);
    ga_main<<<dim3(Mtotal / ROWS, V_), dim3(256), 0, stream>>>(in, kern, S, Bias, out);
}